// AdaptiveLinearWithChannel_81741817577970
// MI455X (gfx1250) — compile-verified
//
#include <hip/hip_runtime.h>

typedef __attribute__((ext_vector_type(2))) float v2f;
typedef __attribute__((ext_vector_type(8))) float v8f;

#define B_    2
#define C_    128
#define N_    16384
#define DIN_  32
#define DOUT_ 32
#define HID_  64

#define WAVES          8
#define PTS_PER_BLOCK  2048

// ---------------------------------------------------------------------------
// Kernel 1: per-channel hyper MLP (1 -> 64 -> 64 -> 32) + bias, producing
// shift[b][c][o] = mlp(t[b]; params[indices[c]])/DIN + bias[c][o]
// One block (64 threads = 2 waves) per channel c.
// ---------------------------------------------------------------------------
__global__ void hyper_shift_kernel(const int* __restrict__ indices,
                                   const float* __restrict__ t,
                                   const float* __restrict__ bias,
                                   const float* __restrict__ hW1,
                                   const float* __restrict__ hb1,
                                   const float* __restrict__ hW2,
                                   const float* __restrict__ hb2,
                                   const float* __restrict__ hW3,
                                   const float* __restrict__ hb3,
                                   float* __restrict__ shift) {
  const int c  = blockIdx.x;
  const int j  = threadIdx.x;           // 0..63
  const int ic = indices[c];

  __shared__ float h1[B_][HID_];
  __shared__ float h2[B_][HID_];

  // layer 1: h1 = relu(t * W1 + b1)
  const float w1 = hW1[ic * HID_ + j];
  const float b1 = hb1[ic * HID_ + j];
  for (int b = 0; b < B_; ++b) {
    float v = t[b] * w1 + b1;
    h1[b][j] = v > 0.f ? v : 0.f;
  }
  __syncthreads();

  // layer 2: h2 = relu(h1 @ W2 + b2)
  const float b2 = hb2[ic * HID_ + j];
  for (int b = 0; b < B_; ++b) {
    float acc = b2;
    for (int k = 0; k < HID_; ++k)
      acc += h1[b][k] * hW2[((size_t)ic * HID_ + k) * HID_ + j];
    h2[b][j] = acc > 0.f ? acc : 0.f;
  }
  __syncthreads();

  // layer 3: out = (h2 @ W3 + b3)/DIN + bias
  if (j < DOUT_) {
    const float b3 = hb3[ic * DOUT_ + j];
    const float bb = bias[c * DOUT_ + j];
    for (int b = 0; b < B_; ++b) {
      float acc = b3;
      for (int k = 0; k < HID_; ++k)
        acc += h2[b][k] * hW3[((size_t)ic * HID_ + k) * DOUT_ + j];
      shift[((size_t)b * C_ + c) * DOUT_ + j] = acc * (1.0f / DIN_) + bb;
    }
  }
}

// ---------------------------------------------------------------------------
// Kernel 2: streaming batched GEMM via V_WMMA_F32_16X16X4_F32.
// Grid: (B*C, N/PTS_PER_BLOCK), block = 256 threads (8 waves).
// Each wave: 16-point strip -> two 16x16 output tiles, K=32 in 8 chunks of 4.
// x is read once and out written once (1 GB total vs 192 MB L2), so both are
// streamed with non-temporal cache policy to avoid L2 thrash.
// ---------------------------------------------------------------------------
__global__ void __launch_bounds__(256)
adaptive_linear_wmma(const float* __restrict__ x,
                     const int* __restrict__ indices,
                     const float* __restrict__ weight,
                     const float* __restrict__ shift,
                     float* __restrict__ out) {
  const int bc   = blockIdx.x;           // b*C + c
  const int c    = bc & (C_ - 1);
  const int lane = threadIdx.x & 31;
  const int wave = threadIdx.x >> 5;
  const int lo   = lane & 15;            // N index within tile / row within strip
  const int hi   = lane >> 4;            // K-half selector per ISA A/B layout

  const float* W = weight + (size_t)indices[c] * (DIN_ * DOUT_);

  // B-matrix fragments, resident for the whole block.
  // Layout (mirror of documented 16x4 A layout): lanes 0-15 carry K=4kc+{0,1},
  // lanes 16-31 carry K=4kc+{2,3}; N = 16*ct + lo.
  v2f wf[2][8];
#pragma unroll
  for (int ct = 0; ct < 2; ++ct)
#pragma unroll
    for (int kc = 0; kc < 8; ++kc) {
      const int k0 = 4 * kc + 2 * hi;
      wf[ct][kc].x = W[(size_t)k0 * DOUT_ + 16 * ct + lo];
      wf[ct][kc].y = W[(size_t)(k0 + 1) * DOUT_ + 16 * ct + lo];
    }

  // Broadcast shift values (bias + hyper) for this (b,c).
  const float sv0 = shift[(size_t)bc * DOUT_ + lo];
  const float sv1 = shift[(size_t)bc * DOUT_ + 16 + lo];

  const size_t rowbase = (size_t)bc * N_;
  const int n_start = blockIdx.y * PTS_PER_BLOCK;

  for (int s = wave; s < PTS_PER_BLOCK / 16; s += WAVES) {
    const int n0 = n_start + s * 16;
    // A fragment source: row = n0+lo, element offset 2*hi within each K-chunk.
    const float* xrow = x + (rowbase + n0 + lo) * DIN_ + 2 * hi;

    v8f acc0 = {};
    v8f acc1 = {};
#pragma unroll
    for (int kc = 0; kc < 8; ++kc) {
      // streaming read: non-temporal (TH_NT) — each x element used exactly once
      v2f a = __builtin_nontemporal_load((const v2f*)(xrow + 4 * kc));
      acc0 = __builtin_amdgcn_wmma_f32_16x16x4_f32(
          false, a, false, wf[0][kc], (short)0, acc0, false, false);
      acc1 = __builtin_amdgcn_wmma_f32_16x16x4_f32(
          false, a, false, wf[1][kc], (short)0, acc1, false, false);
    }

    // D layout: VGPR v holds M = v + 8*hi, N = lo (tile 0) / 16+lo (tile 1).
    float* orow = out + (rowbase + n0 + 8 * hi) * DOUT_;
#pragma unroll
    for (int v = 0; v < 8; ++v) {
      // streaming write: non-temporal — output never re-read
      __builtin_nontemporal_store(acc0[v] + sv0, orow + (size_t)v * DOUT_ + lo);
      __builtin_nontemporal_store(acc1[v] + sv1, orow + (size_t)v * DOUT_ + 16 + lo);
    }
  }
}

// ---------------------------------------------------------------------------
extern "C" void kernel_launch(void* const* d_in, const int* in_sizes, int n_in,
                              void* d_out, int out_size, void* d_ws, size_t ws_size,
                              hipStream_t stream) {
  const float* x       = (const float*)d_in[0];
  const int*   indices = (const int*)d_in[1];
  const float* t       = (const float*)d_in[2];
  const float* weight  = (const float*)d_in[3];
  const float* bias    = (const float*)d_in[4];
  const float* hW1     = (const float*)d_in[5];
  const float* hb1     = (const float*)d_in[6];
  const float* hW2     = (const float*)d_in[7];
  const float* hb2     = (const float*)d_in[8];
  const float* hW3     = (const float*)d_in[9];
  const float* hb3     = (const float*)d_in[10];

  float* shift = (float*)d_ws;          // B*C*DOUT floats = 32 KB
  float* out   = (float*)d_out;

  hyper_shift_kernel<<<C_, HID_, 0, stream>>>(
      indices, t, bias, hW1, hb1, hW2, hb2, hW3, hb3, shift);

  dim3 grid(B_ * C_, N_ / PTS_PER_BLOCK);
  adaptive_linear_wmma<<<grid, 256, 0, stream>>>(x, indices, weight, shift, out);
}